// ART_56573309223834
// MI455X (gfx1250) — compile-verified
//
#include <hip/hip_runtime.h>
#include <math.h>

typedef __attribute__((ext_vector_type(2))) float v2f;
typedef __attribute__((ext_vector_type(4))) float v4f;
typedef __attribute__((ext_vector_type(8))) float v8f;
typedef __attribute__((ext_vector_type(4))) int   v4i;

#define NB 16
#define NA 90
#define ND 256
#define NH 256
#define NW 256
#define NITER 3
#define TWO_PI_F 6.2831853071795864769f
#define INV_C (1.0f / 256.0f)
#define PI_D 3.14159265358979323846

// LDS strides padded to 258 dwords: 258 % 64 == 2 -> lanes m=0..15 hit distinct
// banks, and even stride keeps the v2f (ds_load_b64) operands 8B aligned.
#define MTS 258
#define R0S 258
#define RESS 258
#define MT_FLTS (256 * MTS)          // 66048
#define R0_OFF  MT_FLTS              // 66048
#define RES_OFF (R0_OFF + 16 * R0S)  // 70176
#define COS_OFF (RES_OFF + 16 * RESS)// 74304
#define SIN_OFF (COS_OFF + NA)       // 74394
#define K1_FLTS (SIN_OFF + NA)       // 74484 floats = 297936 B  (<= 320KB LDS)

__device__ __forceinline__ int det_index(float xc, float yc, float c, float s) {
  float rot = fmaf(yc, s, xc * c);
  // faithful to: (rot / (2*pi) * 256).astype(int32) then clamp
  int d = (int)((rot / TWO_PI_F) * 256.0f);
  d = d < 0 ? 0 : (d > 255 ? 255 : d);
  return d;
}

__device__ __forceinline__ void atomicMaxF(float* addr, float v) {
  // sign-aware int-punning max; init cell with -inf bits
  if (v >= 0.0f) atomicMax((int*)addr, __float_as_int(v));
  else           atomicMin((unsigned int*)addr, __float_as_uint(v));
}

// ---------------------------------------------------------------------------
// Phase 1: single-workgroup sequential ART recurrence over 3x90 steps.
// fp(16x256) = r0(16x256) x M_a(256x256) via V_WMMA_F32_16X16X4_F32.
// ---------------------------------------------------------------------------
__global__ void __launch_bounds__(512)
art_seq_kernel(const float* __restrict__ sino, float* __restrict__ Rg,
               float* __restrict__ gmax) {
  extern __shared__ float lds[];
  float* mt   = lds;            // M_a^T : [x][d], 256 x MTS
  float* r0   = lds + R0_OFF;   // row-0 image state: [b][x], 16 x R0S
  float* res  = lds + RES_OFF;  // residual staging:  [b][d], 16 x RESS
  float* cosT = lds + COS_OFF;  // 90
  float* sinT = lds + SIN_OFF;  // 90

  const int tid  = (int)threadIdx.x;
  const int lane = tid & 31;
  const int wave = tid >> 5;    // 16 waves
  const int ln   = lane & 15;
  const int hi   = lane >> 4;   // 0: lanes 0-15, 1: lanes 16-31

  if (tid < NA) {
    double ang = (double)tid * (PI_D / 89.0);   // linspace(0, pi, 90)
    cosT[tid] = (float)cos(ang);
    sinT[tid] = (float)sin(ang);
  }
  for (int i = tid; i < 16 * R0S; i += 512) r0[i] = 0.0f;
  {  // zero residual accumulator in workspace (poisoned by harness)
    v4f z = {0.0f, 0.0f, 0.0f, 0.0f};
    v4f* p = (v4f*)Rg;
    for (int i = tid; i < (NA * NB * ND) / 4; i += 512) p[i] = z;
  }
  if (tid == 0) *gmax = __int_as_float(0xFF800000);  // -inf
  __syncthreads();

  for (int it = 0; it < NITER; ++it) {
    for (int a = 0; a < NA; ++a) {
      const float c = cosT[a];
      const float s = sinT[a];

      // ---- 1) zero M_a^T --------------------------------------------------
      {
        v4f z = {0.0f, 0.0f, 0.0f, 0.0f};
        v4f* p = (v4f*)mt;
        for (int i = tid; i < MT_FLTS / 4; i += 512) p[i] = z;
      }
      __syncthreads();

      // ---- 2) build count matrix via run-length histogram ----------------
      // thread t: column x = t%256, y-half = t/256 (idx monotone in y since
      // sin(theta) >= 0, so runs are contiguous; ds_add_f32 merges halves)
      {
        int x    = tid & 255;
        int half = tid >> 8;
        float xc = (float)x - 128.0f;
        float* row = mt + x * MTS;
        int   prev = -1;
        float runf = 0.0f;
        int y0 = half * 128;
        for (int y = y0; y < y0 + 128; ++y) {
          int d = det_index(xc, (float)y - 128.0f, c, s);
          if (d == prev) {
            runf += 1.0f;
          } else {
            if (prev >= 0) atomicAdd(&row[prev], runf);
            prev = d;
            runf = 1.0f;
          }
        }
        atomicAdd(&row[prev], runf);
      }
      // prefetch next angle's sinogram slice (16 batches x 8 lines)
      {
        int an = (a + 1 < NA) ? a + 1 : 0;
        if (tid < 128) {
          int b = tid >> 3, l8 = tid & 7;
          __builtin_prefetch(&sino[(b * NA + an) * ND + l8 * 32], 0, 3);
        }
      }
      __syncthreads();

      // ---- 3) fp = r0 x M_a  (one 16-col N-tile per wave) ----------------
      // A (16x4 f32): lane L -> m=L%16, VGPR0=K(2*hi), VGPR1=K(2*hi+1)
      // B (4x16 f32): lane L -> n=n0+L%16, VGPR0=K(2*hi), VGPR1=K(2*hi+1)
      const int n0 = wave * 16;
      const float* aBase = r0 + ln * R0S + 2 * hi;
      const float* bBase = mt + (n0 + ln) * MTS + 2 * hi;
      v8f c0 = {0.0f, 0.0f, 0.0f, 0.0f, 0.0f, 0.0f, 0.0f, 0.0f};
      v8f c1 = {0.0f, 0.0f, 0.0f, 0.0f, 0.0f, 0.0f, 0.0f, 0.0f};
#pragma unroll
      for (int k0 = 0; k0 < ND; k0 += 8) {
        v2f av0 = *(const v2f*)(aBase + k0);
        v2f bv0 = *(const v2f*)(bBase + k0);
        c0 = __builtin_amdgcn_wmma_f32_16x16x4_f32(false, av0, false, bv0,
                                                   (short)0, c0, false, false);
        v2f av1 = *(const v2f*)(aBase + k0 + 4);
        v2f bv1 = *(const v2f*)(bBase + k0 + 4);
        c1 = __builtin_amdgcn_wmma_f32_16x16x4_f32(false, av1, false, bv1,
                                                   (short)0, c1, false, false);
      }
      v8f fp = c0 + c1;

      // ---- 4) residual, accumulate per-angle R, stage residual in LDS ----
      // C/D layout: vgpr i -> M=i (lanes 0-15) or M=i+8 (lanes 16-31); N=ln
      {
        int d = n0 + ln;
#pragma unroll
        for (int i = 0; i < 8; ++i) {
          int b = i + 8 * hi;
          float pr = sino[(b * NA + a) * ND + d];
          float r  = pr - fp[i];
          float* gp = &Rg[(a * NB + b) * ND + d];   // unique owner lane
          *gp = *gp + r;
          res[b * RESS + d] = r;
        }
      }
      __syncthreads();

      // ---- 5) row-0 state update: r0[b][x] += res[b][idx_a(0,x)] / 256 ---
      for (int t = tid; t < NB * NW; t += 512) {
        int x = t & 255, b = t >> 8;
        int d0 = det_index((float)x - 128.0f, -128.0f, c, s);
        r0[b * R0S + x] += res[b * RESS + d0] * INV_C;
      }
      __syncthreads();
    }
  }
}

// ---------------------------------------------------------------------------
// Phase 2: back-projection of accumulated residuals + global-max reduction.
// One workgroup per (batch, image row); R[:,b,:] slab staged in LDS (92KB).
// ---------------------------------------------------------------------------
#define K2_FLTS (NA * ND + 2 * NA + 8)

__global__ void __launch_bounds__(256)
art_backproj_kernel(const float* __restrict__ Rg, float* __restrict__ out,
                    float* __restrict__ gmax) {
  extern __shared__ float lds[];
  float* Rb   = lds;                 // [a][d], 90 x 256
  float* cosT = lds + NA * ND;
  float* sinT = cosT + NA;
  float* wred = sinT + NA;           // 8 per-wave maxima

  const int tid = (int)threadIdx.x;
  const int blk = (int)blockIdx.x;
  const int b = blk >> 8;
  const int y = blk & 255;

  if (tid < NA) {
    double ang = (double)tid * (PI_D / 89.0);
    cosT[tid] = (float)cos(ang);
    sinT[tid] = (float)sin(ang);
  }

  // stage R[:, b, :] into LDS (rows strided NB*ND apart in global)
#if __has_builtin(__builtin_amdgcn_global_load_async_to_lds_b128)
  for (int i = tid; i < NA * (ND / 4); i += 256) {
    int a = i >> 6, ch = i & 63;
    const v4f* src = (const v4f*)(Rg + (a * NB + b) * ND) + ch;
    v4f* dst = (v4f*)(Rb + a * ND) + ch;
    __builtin_amdgcn_global_load_async_to_lds_b128(
        (__attribute__((address_space(1))) v4i*)(void*)src,
        (__attribute__((address_space(3))) v4i*)(void*)dst, 0, 0);
  }
  asm volatile("s_wait_asynccnt 0" ::: "memory");
#else
  for (int i = tid; i < NA * (ND / 4); i += 256) {
    int a = i >> 6, ch = i & 63;
    const v4f* src = (const v4f*)(Rg + (a * NB + b) * ND) + ch;
    ((v4f*)(Rb + a * ND))[ch] = *src;
  }
#endif
  __syncthreads();

  const int x = tid;
  const float xc = (float)x - 128.0f;
  const float yc = (float)y - 128.0f;
  float acc = 0.0f;
  for (int a = 0; a < NA; ++a) {
    int d = det_index(xc, yc, cosT[a], sinT[a]);
    acc += Rb[a * ND + d];
  }
  float img = acc * INV_C;
  out[(b * NH + y) * NW + x] = img;

  // global max of the un-clamped image
  float m = img;
#pragma unroll
  for (int off = 16; off > 0; off >>= 1) m = fmaxf(m, __shfl_xor(m, off));
  if ((tid & 31) == 0) wred[tid >> 5] = m;
  __syncthreads();
  if (tid == 0) {
    float mm = wred[0];
#pragma unroll
    for (int i = 1; i < 8; ++i) mm = fmaxf(mm, wred[i]);
    atomicMaxF(gmax, mm);
  }
}

// ---------------------------------------------------------------------------
// Phase 3: out = clip(img, 0, global_max)
// ---------------------------------------------------------------------------
__global__ void __launch_bounds__(256)
art_clamp_kernel(float* __restrict__ out, const float* __restrict__ gmax,
                 int n) {
  int i = (int)(blockIdx.x * blockDim.x + threadIdx.x);
  if (i < n) {
    float g = *gmax;
    float v = out[i];
    out[i] = fminf(fmaxf(v, 0.0f), g);
  }
}

extern "C" void kernel_launch(void* const* d_in, const int* in_sizes, int n_in,
                              void* d_out, int out_size, void* d_ws,
                              size_t ws_size, hipStream_t stream) {
  (void)in_sizes; (void)n_in; (void)out_size; (void)ws_size;
  const float* sino = (const float*)d_in[0];          // (16, 90, 256) f32
  float* out = (float*)d_out;                         // (16, 256, 256) f32
  float* Rg  = (float*)d_ws;                          // 90*16*256 f32 = 1.44MB
  float* gmax = (float*)((char*)d_ws + (size_t)NA * NB * ND * sizeof(float));

  art_seq_kernel<<<1, 512, K1_FLTS * sizeof(float), stream>>>(sino, Rg, gmax);
  art_backproj_kernel<<<NB * NH, 256, K2_FLTS * sizeof(float), stream>>>(
      Rg, out, gmax);
  art_clamp_kernel<<<(NB * NH * NW + 255) / 256, 256, 0, stream>>>(
      out, gmax, NB * NH * NW);
}